// NSFLayer_36206574305599
// MI455X (gfx1250) — compile-verified
//
#include <hip/hip_runtime.h>
#include <hip/hip_bf16.h>
#include <math.h>

// ---------------- constants (match reference) ----------------
#define BATCH   8
#define NSEQ    512
#define EDIM    32
#define HALF    16
#define HID     64
#define KBINS   128
#define NCOEF   383          // 3*K-1
#define OUTDIM  6128         // NCOEF*HALF
#define NTOK    4096         // BATCH*NSEQ
#define BOUND   3.0f
#define MIN_BWH 0.001f
#define MIN_D   0.001f

typedef float v2f __attribute__((ext_vector_type(2)));
typedef float v8f __attribute__((ext_vector_type(8)));

// D = A(16x4,f32) * B(4x16,f32) + C(16x16,f32)
__device__ __forceinline__ v8f wmma4(v2f a, v2f b, v8f c) {
  return __builtin_amdgcn_wmma_f32_16x16x4_f32(false, a, false, b, (short)0, c,
                                               false, false);
}

__device__ __forceinline__ float softplusf(float x) {
  return (x > 20.f) ? x : log1pf(__expf(x));
}

// ================= actnorm + 1x1 conv ========================
// grid: NTOK/16 blocks of 32 threads (1 wave). z = (x*exp(ls)+mu) @ Wc
__global__ void k_actnorm_conv(const float* __restrict__ x,
                               const float* __restrict__ mu,
                               const float* __restrict__ ls,
                               const float* __restrict__ Wc,
                               float* __restrict__ z) {
  __shared__ float As[16][36];
  int lane = threadIdx.x;
  int mrow = lane & 15, hi = lane >> 4;
  const float* xt = x + (size_t)blockIdx.x * 16 * EDIM;
  float* zt = z + (size_t)blockIdx.x * 16 * EDIM;
  for (int i = lane; i < 16 * EDIM; i += 32) {
    int m = i >> 5, c = i & 31;
    As[m][c] = xt[i] * expf(ls[c]) + mu[c];
  }
  __syncthreads();
  for (int nt = 0; nt < 2; ++nt) {
    v8f C = {};
    int n = nt * 16 + mrow;
    for (int k0 = 0; k0 < EDIM; k0 += 4) {
      int kk = k0 + hi * 2;
      v2f a, b;
      a.x = As[mrow][kk];       a.y = As[mrow][kk + 1];
      b.x = Wc[kk * EDIM + n];  b.y = Wc[(kk + 1) * EDIM + n];
      C = wmma4(a, b, C);
    }
    for (int r = 0; r < 8; ++r) zt[(r + 8 * hi) * EDIM + n] = C[r];
  }
}

// ========== log_det base: (sum(ls) + slogdet(Wc)) * v[b] ======
__global__ void k_logdet_base(const float* __restrict__ ls,
                              const float* __restrict__ Wc,
                              const int* __restrict__ v,
                              float* __restrict__ logdet) {
  __shared__ float A[32][33];
  __shared__ float res;
  int t = threadIdx.x;
  for (int i = t; i < 1024; i += 32) A[i >> 5][i & 31] = Wc[i];
  __syncthreads();
  if (t == 0) {
    float lad = 0.f;
    for (int k = 0; k < 32; ++k) {
      int p = k; float mx = fabsf(A[k][k]);
      for (int r = k + 1; r < 32; ++r) {
        float aa = fabsf(A[r][k]);
        if (aa > mx) { mx = aa; p = r; }
      }
      if (p != k)
        for (int c = k; c < 32; ++c) { float tmp = A[k][c]; A[k][c] = A[p][c]; A[p][c] = tmp; }
      float piv = A[k][k];
      lad += logf(fabsf(piv));
      float inv = 1.f / piv;
      for (int r = k + 1; r < 32; ++r) {
        float f = A[r][k] * inv;
        for (int c = k + 1; c < 32; ++c) A[r][c] -= f * A[k][c];
      }
    }
    float sls = 0.f;
    for (int i = 0; i < 32; ++i) sls += ls[i];
    res = sls + lad;
  }
  __syncthreads();
  if (t < BATCH) logdet[t] = res * (float)v[t];
}

// ================= generic tile GEMM =========================
// grid (rows/16, Nc/16), block 32. Y = act(X[rows x K,ldx] @ W[K x Nc] + b)
__global__ void k_linear(const float* __restrict__ X, int ldx,
                         const float* __restrict__ W,
                         const float* __restrict__ bias,
                         float* __restrict__ Y, int ldy, int K, int Nc,
                         int act) {
  int lane = threadIdx.x;
  int mrow = lane & 15, hi = lane >> 4;
  const float* Xt = X + (size_t)blockIdx.x * 16 * ldx;
  int n = blockIdx.y * 16 + mrow;
  v8f C = {};
  for (int k0 = 0; k0 < K; k0 += 4) {
    int kk = k0 + hi * 2;
    v2f a, b;
    a.x = Xt[(size_t)mrow * ldx + kk]; a.y = Xt[(size_t)mrow * ldx + kk + 1];
    b.x = W[(size_t)kk * Nc + n];      b.y = W[(size_t)(kk + 1) * Nc + n];
    C = wmma4(a, b, C);
  }
  float bv = bias[n];
  for (int r = 0; r < 8; ++r) {
    float val = C[r] + bv;
    if (act == 2) val = tanhf(val);
    Y[(size_t)(blockIdx.x * 16 + r + 8 * hi) * ldy + n] = val;
  }
}

// ============ layernorm (+optional residual relu) ============
// grid rows, block 64.  if T: x = x + relu(T);  then LN over 64 feats.
__global__ void k_ln(float* __restrict__ X, const float* __restrict__ g,
                     const float* __restrict__ bta,
                     const float* __restrict__ T) {
  __shared__ float sm[64];
  int t = threadIdx.x;
  size_t off = (size_t)blockIdx.x * HID + t;
  float val = X[off];
  if (T) val += fmaxf(T[off], 0.f);
  sm[t] = val;
  __syncthreads();
  for (int s = 32; s > 0; s >>= 1) { if (t < s) sm[t] += sm[t + s]; __syncthreads(); }
  float mean = sm[0] / 64.f;
  __syncthreads();
  float d = val - mean;
  sm[t] = d * d;
  __syncthreads();
  for (int s = 32; s > 0; s >>= 1) { if (t < s) sm[t] += sm[t + s]; __syncthreads(); }
  float var = sm[0] / 64.f;
  X[off] = d * rsqrtf(var + 1e-5f) * g[t] + bta[t];
}

// ================= MAB0 attention ============================
// grid BATCH, block 256. q shared 16x64; keys/vals (b,512,64); masked softmax.
__global__ void k_attn0(const float* __restrict__ qsg,
                        const float* __restrict__ kb,
                        const float* __restrict__ vb,
                        const int* __restrict__ v, float* __restrict__ o) {
  __shared__ float qsh[16][68];
  __shared__ float lg[16][512];
  __shared__ float red[16][17];
  int tid = threadIdx.x, bb = blockIdx.x;
  int wave = tid >> 5, lane = tid & 31;
  int mrow = lane & 15, hi = lane >> 4;
  for (int i = tid; i < 16 * HID; i += 256) qsh[i >> 6][i & 63] = qsg[i];
  __syncthreads();
  int vcount = v[bb];
  const float* kbB = kb + (size_t)bb * NSEQ * HID;
  const float* vbB = vb + (size_t)bb * NSEQ * HID;
  // logits = q @ k^T / 8, masked
  for (int t = 0; t < 4; ++t) {
    int col = wave * 64 + t * 16 + mrow;
    v8f C = {};
    for (int k0 = 0; k0 < HID; k0 += 4) {
      int kk = k0 + hi * 2;
      v2f a, b;
      a.x = qsh[mrow][kk];             a.y = qsh[mrow][kk + 1];
      b.x = kbB[(size_t)col * HID + kk]; b.y = kbB[(size_t)col * HID + kk + 1];
      C = wmma4(a, b, C);
    }
    bool ok = col < vcount;
    for (int r = 0; r < 8; ++r) lg[r + 8 * hi][col] = ok ? C[r] * 0.125f : -1e9f;
  }
  __syncthreads();
  // row softmax over 512 cols: 16 threads per row
  int row = tid >> 4, sub = tid & 15;
  float mx = -1e30f;
  for (int c = sub; c < 512; c += 16) mx = fmaxf(mx, lg[row][c]);
  red[row][sub] = mx;
  __syncthreads();
  if (sub == 0) {
    float m2 = red[row][0];
    for (int i = 1; i < 16; ++i) m2 = fmaxf(m2, red[row][i]);
    red[row][16] = m2;
  }
  __syncthreads();
  mx = red[row][16];
  float s = 0.f;
  for (int c = sub; c < 512; c += 16) {
    float e = __expf(lg[row][c] - mx);
    lg[row][c] = e; s += e;
  }
  __syncthreads();
  red[row][sub] = s;
  __syncthreads();
  if (sub == 0) {
    float s2 = 0.f;
    for (int i = 0; i < 16; ++i) s2 += red[row][i];
    red[row][16] = s2;
  }
  __syncthreads();
  float inv = 1.f / red[row][16];
  for (int c = sub; c < 512; c += 16) lg[row][c] *= inv;
  __syncthreads();
  // o = q + P @ v
  if (wave < 4) {
    int n = wave * 16 + mrow;
    v8f C = {};
    for (int k0 = 0; k0 < 512; k0 += 4) {
      int kk = k0 + hi * 2;
      v2f a, b;
      a.x = lg[mrow][kk];                a.y = lg[mrow][kk + 1];
      b.x = vbB[(size_t)kk * HID + n];   b.y = vbB[(size_t)(kk + 1) * HID + n];
      C = wmma4(a, b, C);
    }
    for (int r = 0; r < 8; ++r) {
      int m = r + 8 * hi;
      o[((size_t)bb * 16 + m) * HID + n] = C[r] + qsh[m][n];
    }
  }
}

// ================= MAB1 attention ============================
// grid BATCH*(NSEQ/16), block 32. 16 queries, 16 keys, no mask.
__global__ void k_attn1(const float* __restrict__ qb,
                        const float* __restrict__ kb,
                        const float* __restrict__ vb, float* __restrict__ o) {
  __shared__ float P[16][20];
  int lane = threadIdx.x;
  int mrow = lane & 15, hi = lane >> 4;
  int bb = blockIdx.x >> 5, mt = blockIdx.x & 31;
  const float* qt = qb + ((size_t)bb * NSEQ + mt * 16) * HID;
  const float* kt = kb + (size_t)bb * 16 * HID;
  const float* vt = vb + (size_t)bb * 16 * HID;
  v8f C = {};
  for (int k0 = 0; k0 < HID; k0 += 4) {
    int kk = k0 + hi * 2;
    v2f a, b;
    a.x = qt[(size_t)mrow * HID + kk]; a.y = qt[(size_t)mrow * HID + kk + 1];
    b.x = kt[(size_t)mrow * HID + kk]; b.y = kt[(size_t)mrow * HID + kk + 1];
    C = wmma4(a, b, C);
  }
  for (int r = 0; r < 8; ++r) P[r + 8 * hi][mrow] = C[r] * 0.125f;
  __syncthreads();
  if (lane < 16) {
    float mx = -1e30f;
    for (int c = 0; c < 16; ++c) mx = fmaxf(mx, P[lane][c]);
    float s = 0.f;
    for (int c = 0; c < 16; ++c) { float e = __expf(P[lane][c] - mx); P[lane][c] = e; s += e; }
    float inv = 1.f / s;
    for (int c = 0; c < 16; ++c) P[lane][c] *= inv;
  }
  __syncthreads();
  float* ot = o + ((size_t)bb * NSEQ + mt * 16) * HID;
  for (int nt = 0; nt < 4; ++nt) {
    int n = nt * 16 + mrow;
    v8f C2 = {};
    for (int k0 = 0; k0 < 16; k0 += 4) {
      int kk = k0 + hi * 2;
      v2f a, b;
      a.x = P[mrow][kk];                a.y = P[mrow][kk + 1];
      b.x = vt[(size_t)kk * HID + n];   b.y = vt[(size_t)(kk + 1) * HID + n];
      C2 = wmma4(a, b, C2);
    }
    for (int r = 0; r < 8; ++r) {
      int m = r + 8 * hi;
      ot[(size_t)m * HID + n] = C2[r] + qt[(size_t)m * HID + n];
    }
  }
}

// ================= RQS spline evaluation =====================
// cf: 383 coefs = uW[128] | uH[128] | uD[127]  (raw MLP outputs)
__device__ void rqs_eval(const float* cf, float x, float* zo, float* ldo) {
  bool inside = (x >= -BOUND) && (x <= BOUND);
  float xi = fminf(fmaxf(x, -BOUND), BOUND);
  // Wb = 2B*softmax(uW); w = softmax(Wb); same for H.
  float m1w = -1e30f, m1h = -1e30f;
  for (int i = 0; i < KBINS; ++i) {
    m1w = fmaxf(m1w, cf[i]);
    m1h = fmaxf(m1h, cf[KBINS + i]);
  }
  float S1w = 0.f, S1h = 0.f;
  for (int i = 0; i < KBINS; ++i) {
    S1w += __expf(cf[i] - m1w);
    S1h += __expf(cf[KBINS + i] - m1h);
  }
  float m2w = 6.f / S1w, m2h = 6.f / S1h;   // max(2B*softmax) = 6/S1
  float S2w = 0.f, S2h = 0.f;
  for (int i = 0; i < KBINS; ++i) {
    S2w += __expf(6.f * __expf(cf[i] - m1w) / S1w - m2w);
    S2h += __expf(6.f * __expf(cf[KBINS + i] - m1h) / S1h - m2h);
  }
  const float scl = 1.f - MIN_BWH * KBINS;  // 0.872
  float cwL = -BOUND, chL = -BOUND, cwA = 0.f, chA = 0.f;
  float in_cw = 0.f, in_w = 1.f, in_ch = 0.f, in_h = 1.f, in_d = 1.f, in_dp1 = 1.f;
  bool found = false;
  for (int i = 0; i < KBINS; ++i) {
    float wi = MIN_BWH + scl * __expf(6.f * __expf(cf[i] - m1w) / S1w - m2w) / S2w;
    float hh = MIN_BWH + scl * __expf(6.f * __expf(cf[KBINS + i] - m1h) / S1h - m2h) / S2h;
    cwA += wi; chA += hh;
    float cwR = (i == KBINS - 1) ? BOUND : (2.f * BOUND * cwA - BOUND);
    float chR = (i == KBINS - 1) ? BOUND : (2.f * BOUND * chA - BOUND);
    float loc = cwR + ((i == KBINS - 1) ? 1e-6f : 0.f);
    if (!found && xi < loc) {
      found = true;
      in_cw = cwL; in_w = cwR - cwL;
      in_ch = chL; in_h = chR - chL;
      in_d   = (i == 0) ? 1.f
                        : (MIN_D + softplusf(softplusf(cf[2 * KBINS + i - 1])));
      in_dp1 = (i == KBINS - 1) ? 1.f
                        : (MIN_D + softplusf(softplusf(cf[2 * KBINS + i])));
    }
    cwL = cwR; chL = chR;
  }
  float th = (xi - in_cw) / in_w;
  float t1m = th * (1.f - th);
  float dl = in_h / in_w;
  float num = in_h * (dl * th * th + in_d * t1m);
  float den = dl + (in_d + in_dp1 - 2.f * dl) * t1m;
  float outv = in_ch + num / den;
  float omt = 1.f - th;
  float dnum = dl * dl * (in_dp1 * th * th + 2.f * dl * t1m + in_d * omt * omt);
  float lad = logf(dnum) - 2.f * logf(den);
  *zo = inside ? outv : x;
  *ldo = inside ? lad : 0.f;
}

// ============ fused W3 GEMM + RQS forward ====================
// grid NTOK/16, block 256.  Avoids the 100MB coefficient tensor.
__global__ void k_w3_rqs(const float* __restrict__ H2,
                         const float* __restrict__ W3,
                         const float* __restrict__ b3,
                         const float* __restrict__ target,  // stride 32
                         const int* __restrict__ v,
                         float* __restrict__ outB,          // stride 32
                         float* __restrict__ logdet) {
  __shared__ float As[16][68];
  __shared__ float coef[2][16][384];
  __shared__ float sred[32];
  int tid = threadIdx.x;
  int wave = tid >> 5, lane = tid & 31;
  int mrow = lane & 15, hi = lane >> 4;
  const float* Ht = H2 + (size_t)blockIdx.x * 16 * HID;
  for (int i = tid; i < 16 * HID; i += 256) As[i >> 6][i & 63] = Ht[i];
  float acc = 0.f;
  __syncthreads();
  for (int jj = 0; jj < HALF; jj += 2) {
    // GEMM: 2 dims x 24 column tiles over 8 waves
    for (int t2 = wave; t2 < 48; t2 += 8) {
      int jl = t2 / 24, j = jj + jl, ct = t2 % 24;
      int cc = ct * 16 + mrow;
      bool okc = cc < NCOEF;
      const float* Wcol = W3 + (size_t)j * NCOEF;
      v8f C = {};
      for (int k0 = 0; k0 < HID; k0 += 4) {
        int kk = k0 + hi * 2;
        v2f a, b;
        a.x = As[mrow][kk]; a.y = As[mrow][kk + 1];
        b.x = okc ? Wcol[(size_t)kk * OUTDIM + cc] : 0.f;
        b.y = okc ? Wcol[(size_t)(kk + 1) * OUTDIM + cc] : 0.f;
        C = wmma4(a, b, C);
      }
      if (okc) {
        float bv = b3[(size_t)j * NCOEF + cc];
        for (int r = 0; r < 8; ++r) coef[jl][r + 8 * hi][cc] = C[r] + bv;
      }
    }
    __syncthreads();
    // spline: 32 (token, dim) pairs
    if (tid < 32) {
      int m = tid & 15, jl = tid >> 4, j = jj + jl;
      size_t tok = (size_t)blockIdx.x * 16 + m;
      float x = target[tok * EDIM + j];
      float zo, ld;
      rqs_eval(&coef[jl][m][0], x, &zo, &ld);
      outB[tok * EDIM + j] = zo;
      int b = (int)(tok >> 9), n = (int)(tok & 511);
      acc += (n < v[b]) ? ld : 0.f;
    }
    __syncthreads();
  }
  if (tid < 32) sred[tid] = acc;
  __syncthreads();
  if (tid == 0) {
    float s = 0.f;
    for (int i = 0; i < 32; ++i) s += sred[i];
    atomicAdd(&logdet[blockIdx.x >> 5], s);  // 32 tiles per batch
  }
}

// ======================= host side ===========================
extern "C" void kernel_launch(void* const* d_in, const int* in_sizes, int n_in,
                              void* d_out, int out_size, void* d_ws,
                              size_t ws_size, hipStream_t stream) {
  const float* x  = (const float*)d_in[0];
  const int*   v  = (const int*)d_in[1];
  const float* mu = (const float*)d_in[2];
  const float* ls = (const float*)d_in[3];
  const float* Wc = (const float*)d_in[4];
  auto P = [&](int i) { return (const float*)d_in[i]; };
  // param layout (insertion-order flatten):
  // stack: projW, projb, then 4 blocks of [I, mab0(12), mab1(12)]
  const int F1 = 5, F2 = 107, MLP = 209;

  float* ws  = (float*)d_ws;
  float* z   = ws;                    // 4096*32
  float* hA  = z   + NTOK * EDIM;     // 4096*64
  float* hB  = hA  + NTOK * HID;
  float* qb  = hB  + NTOK * HID;
  float* kb  = qb  + NTOK * HID;
  float* vb  = kb  + NTOK * HID;
  float* tb  = vb  + NTOK * HID;
  float* m1h = tb  + NTOK * HID;
  float* m2h = m1h + NTOK * HID;
  float* oS  = m2h + NTOK * HID;      // 128*64
  float* qs  = oS  + BATCH * 16 * HID;// 16*64

  float* out    = (float*)d_out;
  float* logdet = out + (size_t)NTOK * EDIM;

  k_actnorm_conv<<<NTOK / 16, 32, 0, stream>>>(x, mu, ls, Wc, z);
  k_logdet_base<<<1, 32, 0, stream>>>(ls, Wc, v, logdet);

  auto linear = [&](const float* X, int ldx, int rows, const float* W,
                    const float* b, int K, int Nc, float* Y, int ldy, int act) {
    dim3 g(rows / 16, Nc / 16);
    k_linear<<<g, 32, 0, stream>>>(X, ldx, W, b, Y, ldy, K, Nc, act);
  };
  auto ln = [&](float* X, int rows, const float* g, const float* b,
                const float* T) {
    k_ln<<<rows, 64, 0, stream>>>(X, g, b, T);
  };

  auto coupling = [&](const float* cond, int SB, const float* target,
                      float* outHalf) {
    float* hcur = hA;
    float* hnxt = hB;
    linear(cond, EDIM, NTOK, P(SB), P(SB + 1), HALF, HID, hcur, HID, 0);
    for (int blk = 0; blk < 4; ++blk) {
      int bb0 = SB + 2 + blk * 25;
      const float* I = P(bb0);
      int m0 = bb0 + 1, m1 = bb0 + 13;
      // ---- mab0: Q = I (shared over batch), Kx = h ----
      linear(I, HID, 16, P(m0 + 0), P(m0 + 1), HID, HID, qs, HID, 0);      // Wq
      linear(hcur, HID, NTOK, P(m0 + 2), P(m0 + 3), HID, HID, kb, HID, 0); // Wk
      linear(hcur, HID, NTOK, P(m0 + 4), P(m0 + 5), HID, HID, vb, HID, 0); // Wv
      k_attn0<<<BATCH, 256, 0, stream>>>(qs, kb, vb, v, oS);
      ln(oS, BATCH * 16, P(m0 + 8), P(m0 + 9), nullptr);                   // ln0
      linear(oS, HID, BATCH * 16, P(m0 + 6), P(m0 + 7), HID, HID, tb, HID, 0); // Wo
      ln(oS, BATCH * 16, P(m0 + 10), P(m0 + 11), tb);                      // +relu, ln1
      // ---- mab1: Q = h, Kx = Hh(oS) ----
      linear(hcur, HID, NTOK, P(m1 + 0), P(m1 + 1), HID, HID, qb, HID, 0);
      linear(oS, HID, BATCH * 16, P(m1 + 2), P(m1 + 3), HID, HID, kb, HID, 0);
      linear(oS, HID, BATCH * 16, P(m1 + 4), P(m1 + 5), HID, HID, vb, HID, 0);
      k_attn1<<<BATCH * (NSEQ / 16), 32, 0, stream>>>(qb, kb, vb, hnxt);
      ln(hnxt, NTOK, P(m1 + 8), P(m1 + 9), nullptr);
      linear(hnxt, HID, NTOK, P(m1 + 6), P(m1 + 7), HID, HID, tb, HID, 0);
      ln(hnxt, NTOK, P(m1 + 10), P(m1 + 11), tb);
      float* t = hcur; hcur = hnxt; hnxt = t;
    }
    // ---- MLP + fused spline ----
    linear(hcur, HID, NTOK, P(MLP + 0), P(MLP + 1), HID, HID, m1h, HID, 2);
    linear(m1h, HID, NTOK, P(MLP + 2), P(MLP + 3), HID, HID, m2h, HID, 2);
    k_w3_rqs<<<NTOK / 16, 256, 0, stream>>>(m2h, P(MLP + 4), P(MLP + 5),
                                            target, v, outHalf, logdet);
  };

  // coupling 1: cond = lower(z[...,:16]) -> transforms upper -> out[...,16:32]
  coupling(z, F1, z + HALF, out + HALF);
  // coupling 2: cond = new upper (out[...,16:32]) -> transforms lower -> out[...,:16]
  coupling(out + HALF, F2, z, out);
}